// CNNLSTMEncoder_50319836840609
// MI455X (gfx1250) — compile-verified
//
#include <hip/hip_runtime.h>

typedef __attribute__((ext_vector_type(2))) float v2f;
typedef __attribute__((ext_vector_type(8))) float v8f;

#define LATENT    64
#define NFACT     10
#define BATCH     64
#define SEQ       4096
#define GATES     256   // 4*LATENT

// ---------------------------------------------------------------------------
// D = A(16x4, f32) * B(4x16, f32) + C(16x16, f32)  -- V_WMMA_F32_16X16X4_F32
// A layout: lane l holds rows M=l&15; VGPR0/1 = K = 2*(l>>4) + {0,1}
// B layout: lane l holds col  N=l&15; VGPR0/1 = K = 2*(l>>4) + {0,1}
// D layout: lane l, vgpr r -> M = r + 8*(l>>4), N = l&15
// ---------------------------------------------------------------------------
__device__ __forceinline__ v8f wmma_f32(v2f a, v2f b, v8f c) {
  return __builtin_amdgcn_wmma_f32_16x16x4_f32(false, a, false, b,
                                               (short)0, c, false, false);
}

__device__ __forceinline__ float fast_sigmoid(float x) {
  return 1.0f / (1.0f + __expf(-x));
}
__device__ __forceinline__ float fast_tanh(float x) {
  float e = __expf(2.0f * x);
  return (e - 1.0f) / (e + 1.0f);
}

// Async DMA: 16 bytes per lane, global -> LDS, tracked by ASYNCcnt.
// ldsoff = byte offset within the workgroup's LDS allocation (addr[31:0]).
__device__ __forceinline__ void async_copy_b128(unsigned ldsoff,
                                                const float* gaddr) {
  asm volatile("global_load_async_to_lds_b128 %0, %1, off"
               :: "v"(ldsoff), "v"(gaddr) : "memory");
}

// ---------------------------------------------------------------------------
// Conv1d (kernel=3, SAME) + bias + ReLU as 3 accumulated shifted WMMA GEMMs.
// One WG = 128 thr (4 waves): one batch row, 16 seq positions, 64 out chans.
// ---------------------------------------------------------------------------
template<int CIN>
__global__ __launch_bounds__(128) void conv_relu_kernel(
    const float* __restrict__ x,    // [B, S, CIN]
    const float* __restrict__ w,    // [3, CIN, 64]  (flax WIO)
    const float* __restrict__ bias, // [64]
    float* __restrict__ y)          // [B, S, 64]
{
  __shared__ float xtile[18 * CIN];            // rows s0-1 .. s0+16
  const int wg  = blockIdx.x;                  // B*S/16 workgroups
  const int b   = wg / (SEQ / 16);
  const int s0  = (wg % (SEQ / 16)) * 16;
  const int tid = threadIdx.x;

  for (int idx = tid; idx < 18 * CIN; idx += 128) {
    int row = idx / CIN;
    int col = idx - row * CIN;
    int s   = s0 - 1 + row;
    xtile[idx] = (s >= 0 && s < SEQ) ? x[(b * SEQ + s) * CIN + col] : 0.0f;
  }
  __syncthreads();

  const int lane = tid & 31;
  const int mr   = lane & 15;           // A row index
  const int kh   = (lane >> 4) * 2;     // K half select
  const int co0  = (tid >> 5) * 16;     // wave -> output-channel tile
  const int n    = lane & 15;
  const int mhi  = (lane >> 4) << 3;

  v8f acc = {};
  #pragma unroll
  for (int ta = 0; ta < 3; ++ta) {
    #pragma unroll
    for (int kb = 0; kb < CIN / 4; ++kb) {
      v2f a = *(const v2f*)&xtile[(mr + ta) * CIN + kb * 4 + kh];   // ds_load_b64
      const float* wp = &w[((ta * CIN + kb * 4 + kh) * 64) + co0 + n];
      v2f bf = { wp[0], wp[64] };
      acc = wmma_f32(a, bf, acc);
    }
  }
  float bb = bias[co0 + n];
  #pragma unroll
  for (int r = 0; r < 8; ++r) {
    int mm = r + mhi;
    y[((size_t)b * SEQ + s0 + mm) * 64 + co0 + n] = fmaxf(acc[r] + bb, 0.0f);
  }
}

// ---------------------------------------------------------------------------
// LSTM input projection: xw[s,b,:] = h[b,s,:] @ Wx + bias  (bias folded here).
// Output is TIME-MAJOR [S, B, 256] so each recurrent step reads 64KB contig.
// One WG = 512 thr (16 waves): one 16-row M tile x all 16 N tiles.
// ---------------------------------------------------------------------------
__global__ __launch_bounds__(512) void lstm_input_proj_kernel(
    const float* __restrict__ h,    // [B*S, 64]
    const float* __restrict__ Wx,   // [64, 256]
    const float* __restrict__ bias, // [256]
    float* __restrict__ xwT)        // [S, B, 256]
{
  __shared__ float htile[16 * 64];
  const int mt  = blockIdx.x;
  const int r0  = mt * 16;
  const int b   = r0 / SEQ;
  const int s0  = r0 % SEQ;
  const int tid = threadIdx.x;

  for (int idx = tid; idx < 16 * 64; idx += 512)
    htile[idx] = h[(size_t)r0 * 64 + idx];
  __syncthreads();

  const int lane = tid & 31;
  const int mr   = lane & 15;
  const int kh   = (lane >> 4) * 2;
  const int n0   = (tid >> 5) * 16;   // wave -> gate-column tile
  const int n    = lane & 15;
  const int mhi  = (lane >> 4) << 3;

  v8f acc = {};
  #pragma unroll
  for (int kb = 0; kb < 16; ++kb) {
    v2f a = *(const v2f*)&htile[mr * 64 + kb * 4 + kh];
    const float* wp = &Wx[(kb * 4 + kh) * GATES + n0 + n];
    v2f bf = { wp[0], wp[GATES] };
    acc = wmma_f32(a, bf, acc);
  }
  float bb = bias[n0 + n];
  #pragma unroll
  for (int r = 0; r < 8; ++r) {
    int mm = r + mhi;
    xwT[((size_t)(s0 + mm) * BATCH + b) * GATES + n0 + n] = acc[r] + bb;
  }
}

// ---------------------------------------------------------------------------
// LSTM recurrence. 4 WGs, one per 16-batch slice (batch rows independent).
// 512 thr = 16 waves; wave w owns gate n-tile w. Wh slab resident in VGPRs
// (bw[16] = 32 VGPRs) for all 4096 steps. h in LDS, c in registers, gates
// round-trip through LDS for the cross-wave i/f/g/o combine. K accumulation
// split into two WMMA chains to halve RAW latency. The next step's 16KB
// activation block is DMA'd into an LDS double buffer with
// GLOBAL_LOAD_ASYNC_TO_LDS_B128, overlapped with the WMMA chain, so no
// global-memory latency sits on the 4096-step critical path.
// ---------------------------------------------------------------------------
__global__ __launch_bounds__(512) void lstm_recurrent_kernel(
    const float* __restrict__ xwT,   // [S, B, 256] (bias pre-added)
    const float* __restrict__ Wh,    // [64, 256]
    float* __restrict__ hout,        // [B, S, 64] (used iff write_h)
    float* __restrict__ cfinal,      // [B, 64]
    int write_h)
{
  __shared__ float hbuf[16 * 64];        //  4 KB current hidden state
  __shared__ float gbuf[16 * 256];       // 16 KB activated gates i|f|g|o
  __shared__ float xwbuf[2][16 * 256];   // 32 KB async double buffer

  const int b0   = blockIdx.x * 16;
  const int tid  = threadIdx.x;
  const int wave = tid >> 5;          // gate n-tile 0..15
  const int lane = tid & 31;
  const int mr   = lane & 15;
  const int kh   = (lane >> 4) * 2;
  const int n0   = wave * 16;
  const int n    = lane & 15;
  const int mhi  = (lane >> 4) << 3;
  const int gate = wave >> 2;         // 0:i 1:f 2:g 3:o

  // LDS byte offsets of the double buffer (addr[31:0] of the generic pointer)
  const unsigned xwlds = (unsigned)(uintptr_t)(&xwbuf[0][0]);

  // Wh fragments for this wave's 16 gate columns: live in VGPRs all steps.
  v2f bw[16];
  #pragma unroll
  for (int kb = 0; kb < 16; ++kb) {
    const float* wp = &Wh[(kb * 4 + kh) * GATES + n0 + n];
    bw[kb] = { wp[0], wp[GATES] };
  }

  for (int idx = tid; idx < 16 * 64; idx += 512) hbuf[idx] = 0.0f;
  const int um = tid >> 6;            // this thread updates rows um, um+8
  const int uj = tid & 63;            // column within LATENT
  float c0 = 0.0f, c1 = 0.0f;

  // Preload step 0's activation block (2 x b128 per thread = 16 KB total).
  {
    const float* g = xwT + ((size_t)0 * BATCH + b0) * GATES;
    async_copy_b128(xwlds + tid * 16u,           g + tid * 4);
    async_copy_b128(xwlds + (tid + 512u) * 16u,  g + (tid + 512) * 4);
  }
  __syncthreads();

  for (int t = 0; t < SEQ; ++t) {
    // Kick off DMA for step t+1 into the other buffer, then overlap with WMMA.
    if (t + 1 < SEQ) {
      const float* g = xwT + ((size_t)(t + 1) * BATCH + b0) * GATES;
      unsigned lb = xwlds + ((unsigned)(t + 1) & 1u) * (16u * 256u * 4u);
      async_copy_b128(lb + tid * 16u,          g + tid * 4);
      async_copy_b128(lb + (tid + 512u) * 16u, g + (tid + 512) * 4);
    }

    v8f accA = {}, accB = {};
    #pragma unroll
    for (int kb = 0; kb < 16; kb += 2) {
      v2f a0 = *(const v2f*)&hbuf[mr * 64 + kb * 4 + kh];
      v2f a1 = *(const v2f*)&hbuf[mr * 64 + (kb + 1) * 4 + kh];
      accA = wmma_f32(a0, bw[kb],     accA);
      accB = wmma_f32(a1, bw[kb + 1], accB);
    }

    // Step t's loads precede step t+1's in ASYNCcnt order: <=2 outstanding
    // means t's data landed (this wave); barrier extends that to all waves.
    if (t + 1 < SEQ) asm volatile("s_wait_asynccnt 0x2" ::: "memory");
    else             asm volatile("s_wait_asynccnt 0x0" ::: "memory");
    __syncthreads();

    const float* xwt = &xwbuf[t & 1][0];
    #pragma unroll
    for (int r = 0; r < 8; ++r) {
      int mm = r + mhi;
      float v = accA[r] + accB[r] + xwt[mm * GATES + n0 + n];
      v = (gate == 2) ? fast_tanh(v) : fast_sigmoid(v);
      gbuf[mm * GATES + n0 + n] = v;
    }
    __syncthreads();

    // c/h update: thread owns (um, uj) and (um+8, uj)
    {
      float i0 = gbuf[um * GATES + uj];
      float f0 = gbuf[um * GATES + 64 + uj];
      float g0 = gbuf[um * GATES + 128 + uj];
      float o0 = gbuf[um * GATES + 192 + uj];
      c0 = f0 * c0 + i0 * g0;
      float h0v = o0 * fast_tanh(c0);
      hbuf[um * 64 + uj] = h0v;

      int m1 = um + 8;
      float i1 = gbuf[m1 * GATES + uj];
      float f1 = gbuf[m1 * GATES + 64 + uj];
      float g1 = gbuf[m1 * GATES + 128 + uj];
      float o1 = gbuf[m1 * GATES + 192 + uj];
      c1 = f1 * c1 + i1 * g1;
      float h1v = o1 * fast_tanh(c1);
      hbuf[m1 * 64 + uj] = h1v;

      if (write_h) {
        hout[((size_t)(b0 + um) * SEQ + t) * 64 + uj] = h0v;
        hout[((size_t)(b0 + m1) * SEQ + t) * 64 + uj] = h1v;
      }
    }
    __syncthreads();
  }

  cfinal[(b0 + um)     * 64 + uj] = c0;
  cfinal[(b0 + um + 8) * 64 + uj] = c1;
}

// ---------------------------------------------------------------------------
// Dense head: out[64,10] = c[64,64] @ dense_w[64,10] + dense_b. Trivial size.
// ---------------------------------------------------------------------------
__global__ __launch_bounds__(640) void dense_head_kernel(
    const float* __restrict__ c,   // [64, 64]
    const float* __restrict__ dw,  // [64, 10]
    const float* __restrict__ db,  // [10]
    float* __restrict__ out)       // [64, 10]
{
  int tid = threadIdx.x;
  if (tid >= BATCH * NFACT) return;
  int row = tid / NFACT, o = tid % NFACT;
  float acc = db[o];
  #pragma unroll
  for (int k = 0; k < LATENT; ++k)
    acc += c[row * LATENT + k] * dw[k * NFACT + o];
  out[tid] = acc;
}

// ---------------------------------------------------------------------------
extern "C" void kernel_launch(void* const* d_in, const int* in_sizes, int n_in,
                              void* d_out, int out_size, void* d_ws, size_t ws_size,
                              hipStream_t stream) {
  const float* x   = (const float*)d_in[0];
  const float* cw0 = (const float*)d_in[1];
  const float* cb0 = (const float*)d_in[2];
  const float* cw1 = (const float*)d_in[3];
  const float* cb1 = (const float*)d_in[4];
  const float* cw2 = (const float*)d_in[5];
  const float* cb2 = (const float*)d_in[6];
  const float* Wx0 = (const float*)d_in[7];
  const float* Wh0 = (const float*)d_in[8];
  const float* b0  = (const float*)d_in[9];
  const float* Wx1 = (const float*)d_in[10];
  const float* Wh1 = (const float*)d_in[11];
  const float* b1  = (const float*)d_in[12];
  const float* dw  = (const float*)d_in[13];
  const float* db  = (const float*)d_in[14];
  float* out = (float*)d_out;

  char*  ws   = (char*)d_ws;
  float* bufA = (float*)(ws);                          //  64 MiB: conv ping / c_final
  float* bufB = (float*)(ws + ((size_t)64  << 20));    //  64 MiB: conv pong / h-seq
  float* xw   = (float*)(ws + ((size_t)128 << 20));    // 256 MiB: [S,B,256] gate acts

  const int mtiles = BATCH * SEQ / 16;                 // 16384

  conv_relu_kernel<128><<<mtiles, 128, 0, stream>>>(x,    cw0, cb0, bufA);
  conv_relu_kernel<64> <<<mtiles, 128, 0, stream>>>(bufA, cw1, cb1, bufB);
  conv_relu_kernel<64> <<<mtiles, 128, 0, stream>>>(bufB, cw2, cb2, bufA);

  // Layer 0: project (bias folded), run recurrence, emit full h sequence.
  lstm_input_proj_kernel<<<mtiles, 512, 0, stream>>>(bufA, Wx0, b0, xw);
  lstm_recurrent_kernel<<<BATCH / 16, 512, 0, stream>>>(xw, Wh0, bufB, bufA, 1);

  // Layer 1: only the final cell state matters downstream.
  lstm_input_proj_kernel<<<mtiles, 512, 0, stream>>>(bufB, Wx1, b1, xw);
  lstm_recurrent_kernel<<<BATCH / 16, 512, 0, stream>>>(xw, Wh1, bufB, bufA, 0);

  dense_head_kernel<<<1, 640, 0, stream>>>(bufA, dw, db, out);
}